// Critic_53584011984901
// MI455X (gfx1250) — compile-verified
//
#include <hip/hip_runtime.h>
#include <hip/hip_bf16.h>

#define BB   128
#define NN   2048
#define HH   128
#define TPB  256
#define CPT  (NN / TPB)   // 8 columns per thread

typedef __attribute__((ext_vector_type(2))) float v2f;
typedef __attribute__((ext_vector_type(8))) float v8f;

// ---------------- fast tanh: native v_tanh_f32 on gfx1250 if available ----
#if __has_builtin(__builtin_amdgcn_tanhf)
__device__ __forceinline__ float fast_tanh(float x) { return __builtin_amdgcn_tanhf(x); }
#else
__device__ __forceinline__ float fast_tanh(float x) {
  // tanh(x) = sign(x) * (1-e)/(1+e), e = 2^(-2|x|*log2(e)); arg<=0 so v_exp_f32 is exact-range
  float ax = __builtin_fabsf(x);
  float e  = __builtin_amdgcn_exp2f(ax * -2.8853900817779268f);
  float t  = (1.0f - e) * __builtin_amdgcn_rcpf(1.0f + e);
  return __builtin_copysignf(t, x);
}
#endif

// ---------------- stage 0: low-rank collapse precompute + hy zero ---------
// pre[i] layout: a0,a1,ud,uc,cE,cT each HH floats (stride 6*HH per block i)
__global__ __launch_bounds__(HH) void precompute_kernel(
    const float* __restrict__ Wd, const float* __restrict__ bd,
    const float* __restrict__ Wc, const float* __restrict__ bc,
    const float* __restrict__ Wr, const float* __restrict__ br,
    const float* __restrict__ W_dyn, const float* __restrict__ b_dyn,
    const float* __restrict__ W_st,  const float* __restrict__ b_st,
    const float* __restrict__ W_ch,  const float* __restrict__ b_ch,
    float* __restrict__ pre, float* __restrict__ hy) {
  const int i = blockIdx.x;
  const int h = threadIdx.x;
  const float* wr = Wr + (i * HH + h) * HH;
  const float* wd = Wd + (i * HH + h) * HH;
  const float* wc = Wc + (i * HH + h) * HH;
  float a0 = 0.f, a1 = 0.f, ud = 0.f, uc = 0.f, ce = 0.f, cd = 0.f;
  for (int k = 0; k < HH; ++k) {
    float wrk = wr[k], wdk = wd[k], wck = wc[k];
    a0 = fmaf(wrk, W_st[2 * k + 0], a0);
    a1 = fmaf(wrk, W_st[2 * k + 1], a1);
    ud = fmaf(wdk, W_dyn[k], ud);
    uc = fmaf(wck, W_ch[k], uc);
    ce = fmaf(wrk, b_st[k], fmaf(wck, b_ch[k], ce));
    cd = fmaf(wdk, b_dyn[k], cd);
  }
  ce += br[i * HH + h] + bc[i * HH + h];
  float ct = ce + cd + bd[i * HH + h];
  float* p = pre + i * 6 * HH;
  p[0 * HH + h] = a0; p[1 * HH + h] = a1; p[2 * HH + h] = ud;
  p[3 * HH + h] = uc; p[4 * HH + h] = ce; p[5 * HH + h] = ct;
  if (i == 0) {  // zero hy for the first q-GEMM (stream-ordered before it)
    for (int j = 0; j < HH; ++j) hy[h * HH + j] = 0.f;
  }
}

// ---------------- WMMA f32 128x128x128 GEMM: D = act(A @ W^T + bias) ------
// D[m][n] = sum_k A[m][k] * W[n][k].  One wave per 16x16 tile, K steps of 4.
// A-frag (16x4 f32, 2 VGPR): lanes 0-15 -> M=l16, K=kk+0/kk+1; lanes 16-31 -> K=kk+2/kk+3.
// B-frag (4x16): mirrored striping (N=l16 per half, VGPR0/1 = consecutive K).
__global__ __launch_bounds__(32) void gemm128_wmma(
    const float* __restrict__ A, const float* __restrict__ W,
    const float* __restrict__ bias, float* __restrict__ D, int relu) {
  const int lane = threadIdx.x;
  const int half = lane >> 4;
  const int l16  = lane & 15;
  const int tm = blockIdx.x * 16;
  const int tn = blockIdx.y * 16;
  const float* Arow = A + (tm + l16) * HH;
  const float* Wrow = W + (tn + l16) * HH;
  v8f acc = {};
#pragma unroll
  for (int kk = 0; kk < HH; kk += 4) {
    const int k = kk + 2 * half;
    v2f a = { Arow[k], Arow[k + 1] };
    v2f b = { Wrow[k], Wrow[k + 1] };
    acc = __builtin_amdgcn_wmma_f32_16x16x4_f32(
        false, a, false, b, (short)0, acc, false, false);
  }
  const int n = tn + l16;
  const float bv = bias[n];
#pragma unroll
  for (int j = 0; j < 8; ++j) {
    const int m = tm + j + 8 * half;
    float val = acc[j] + bv;
    if (relu) val = fmaxf(val, 0.f);
    D[m * HH + n] = val;
  }
}

// ---------------- stage i: attention block (one workgroup per batch b) ----
__global__ __launch_bounds__(TPB) void attn_block_kernel(
    const float* __restrict__ s_static,   // [B,2,N]
    const float* __restrict__ s_dyn,      // [B,1,N]
    const float* __restrict__ s_ch,       // [B,1,N]
    const float* __restrict__ pre,        // this block's 6*HH vectors
    const float* __restrict__ q,          // [B,H]
    const float* __restrict__ vvec,       // [H]
    float* __restrict__ hy) {             // [B,H] (overwritten)
  __shared__ float a0s[HH], a1s[HH], uds[HH], ucs[HH], offs[HH], vs[HH], ces[HH];
  __shared__ float red[TPB / 32];
  __shared__ float red4[TPB / 32][4];
  __shared__ float totals[5];
  const int b = blockIdx.x;
  const int tid = threadIdx.x;
  if (tid < HH) {
    a0s[tid] = pre[0 * HH + tid];
    a1s[tid] = pre[1 * HH + tid];
    uds[tid] = pre[2 * HH + tid];
    ucs[tid] = pre[3 * HH + tid];
    ces[tid] = pre[4 * HH + tid];
    offs[tid] = pre[5 * HH + tid] + q[b * HH + tid];
    vs[tid] = vvec[tid];
  }
  __syncthreads();

  float s0[CPT], s1[CPT], sd[CPT], sc[CPT], u[CPT];
  const float* st = s_static + b * 2 * NN;
#pragma unroll
  for (int c = 0; c < CPT; ++c) {
    const int n = tid + c * TPB;
    s0[c] = st[n];
    s1[c] = st[NN + n];
    sd[c] = s_dyn[b * NN + n];
    sc[c] = s_ch[b * NN + n];
    u[c] = 0.f;
  }
  // u[b,n] = sum_h v[h] * tanh(a0 s0 + a1 s1 + ud sd + uc sc + off[h])
  for (int h = 0; h < HH; ++h) {
    const float A0 = a0s[h], A1 = a1s[h], UD = uds[h], UC = ucs[h];
    const float OFF = offs[h], V = vs[h];
#pragma unroll
    for (int c = 0; c < CPT; ++c) {
      float arg = fmaf(A0, s0[c], fmaf(A1, s1[c],
                  fmaf(UD, sd[c], fmaf(UC, sc[c], OFF))));
      u[c] = fmaf(V, fast_tanh(arg), u[c]);
    }
  }
  // block max (wave32 shuffle + LDS)
  float m = u[0];
#pragma unroll
  for (int c = 1; c < CPT; ++c) m = fmaxf(m, u[c]);
  for (int o = 16; o > 0; o >>= 1) m = fmaxf(m, __shfl_xor(m, o, 32));
  if ((tid & 31) == 0) red[tid >> 5] = m;
  __syncthreads();
  if (tid == 0) {
    float mm = red[0];
    for (int w = 1; w < TPB / 32; ++w) mm = fmaxf(mm, red[w]);
    totals[0] = mm;
  }
  __syncthreads();
  const float gmax = totals[0];
  const float L2E = 1.4426950408889634f;
  float Z = 0.f, P0 = 0.f, P1 = 0.f, Pc = 0.f;
#pragma unroll
  for (int c = 0; c < CPT; ++c) {
    float w = __builtin_amdgcn_exp2f((u[c] - gmax) * L2E);
    Z += w; P0 += w * s0[c]; P1 += w * s1[c]; Pc += w * sc[c];
  }
  for (int o = 16; o > 0; o >>= 1) {
    Z  += __shfl_xor(Z,  o, 32);
    P0 += __shfl_xor(P0, o, 32);
    P1 += __shfl_xor(P1, o, 32);
    Pc += __shfl_xor(Pc, o, 32);
  }
  if ((tid & 31) == 0) {
    red4[tid >> 5][0] = Z;  red4[tid >> 5][1] = P0;
    red4[tid >> 5][2] = P1; red4[tid >> 5][3] = Pc;
  }
  __syncthreads();
  if (tid < 4) {
    float s = 0.f;
    for (int w = 0; w < TPB / 32; ++w) s += red4[w][tid];
    totals[1 + tid] = s;
  }
  __syncthreads();
  if (tid < HH) {
    const float invZ = __builtin_amdgcn_rcpf(totals[1]);
    const float p0 = totals[2] * invZ, p1 = totals[3] * invZ, pc = totals[4] * invZ;
    hy[b * HH + tid] = fmaf(a0s[tid], p0,
                       fmaf(a1s[tid], p1,
                       fmaf(ucs[tid], pc, ces[tid])));
  }
}

// ---------------- head stage 2: out[b] = z1[b,:] . W2 + b2 ----------------
__global__ __launch_bounds__(BB) void head2_kernel(
    const float* __restrict__ z1, const float* __restrict__ W2,
    const float* __restrict__ b2, float* __restrict__ out) {
  const int b = threadIdx.x;
  const float* row = z1 + b * HH;
  float s = 0.f;
  for (int h = 0; h < HH; ++h) s = fmaf(row[h], W2[h], s);
  out[b] = s + b2[0];
}

extern "C" void kernel_launch(void* const* d_in, const int* in_sizes, int n_in,
                              void* d_out, int out_size, void* d_ws, size_t ws_size,
                              hipStream_t stream) {
  const float* state_init  = (const float*)d_in[0];
  const float* s_static    = (const float*)d_in[1];
  const float* static_ch_l = (const float*)d_in[2];
  const float* W_dyn = (const float*)d_in[3];
  const float* b_dyn = (const float*)d_in[4];
  const float* W_st  = (const float*)d_in[5];
  const float* b_st  = (const float*)d_in[6];
  const float* W_ch  = (const float*)d_in[7];
  const float* b_ch  = (const float*)d_in[8];
  const float* v     = (const float*)d_in[9];
  const float* Wd    = (const float*)d_in[10];
  const float* bd    = (const float*)d_in[11];
  const float* Wc    = (const float*)d_in[12];
  const float* bc    = (const float*)d_in[13];
  const float* Wr    = (const float*)d_in[14];
  const float* br    = (const float*)d_in[15];
  const float* Wq    = (const float*)d_in[16];
  const float* bq    = (const float*)d_in[17];
  const float* W1    = (const float*)d_in[18];
  const float* b1    = (const float*)d_in[19];
  const float* W2    = (const float*)d_in[20];
  const float* b2    = (const float*)d_in[21];
  float* out = (float*)d_out;

  float* ws  = (float*)d_ws;
  float* hy  = ws;                 // B*H
  float* q   = ws + BB * HH;       // B*H
  float* z1  = ws + 2 * BB * HH;   // B*H
  float* pre = ws + 3 * BB * HH;   // 3 * 6 * H

  precompute_kernel<<<3, HH, 0, stream>>>(Wd, bd, Wc, bc, Wr, br,
                                          W_dyn, b_dyn, W_st, b_st, W_ch, b_ch,
                                          pre, hy);
  dim3 gtiles(HH / 16, HH / 16);
  for (int i = 0; i < 3; ++i) {
    gemm128_wmma<<<gtiles, 32, 0, stream>>>(hy, Wq + i * HH * HH, bq + i * HH, q, 0);
    attn_block_kernel<<<BB, TPB, 0, stream>>>(s_static, state_init, static_ch_l,
                                              pre + i * 6 * HH, q, v + i * HH, hy);
  }
  gemm128_wmma<<<gtiles, 32, 0, stream>>>(hy, W1, b1, z1, 1);
  head2_kernel<<<1, BB, 0, stream>>>(z1, W2, b2, out);
}